// SimulatorObjectSM_12970801234129
// MI455X (gfx1250) — compile-verified
//
#include <hip/hip_runtime.h>
#include <hip/hip_bf16.h>
#include <math.h>

typedef __attribute__((ext_vector_type(2))) float v2f;
typedef __attribute__((ext_vector_type(8))) float v8f;

// k -> unique-degree index (l=0:1, l=2:5, l=4:9, l=6:13, l=8:17 entries)
__constant__ int c_lidx[45] = {
    0,
    1,1,1,1,1,
    2,2,2,2,2,2,2,2,2,
    3,3,3,3,3,3,3,3,3,3,3,3,3,
    4,4,4,4,4,4,4,4,4,4,4,4,4,4,4,4,4
};

// ---------------------------------------------------------------------------
// Kernel 1: precompute B fragments  B[k][n] = w(t_{4c+k}) * P_{2n}(t_{4c+k})
// stored per (chunk, lane) as float2 matching the 16x16x4 f32 B-operand layout:
//   lanes 0-15  -> col N=lane,    v0 = K=0 row, v1 = K=1 row
//   lanes 16-31 -> col N=lane-16, v0 = K=2 row, v1 = K=3 row
// Columns N>=5 are zero padding.
// ---------------------------------------------------------------------------
__global__ void k_precompute_B(const int* __restrict__ num_t_p,
                               float* __restrict__ Bws) {
    int tid  = blockIdx.x * blockDim.x + threadIdx.x;
    int lane = tid & 31;
    int c    = tid >> 5;
    int num_t  = *num_t_p;
    int chunks = (num_t + 3) >> 2;
    if (c >= chunks) return;

    int   Ncol  = lane & 15;
    int   khalf = (lane < 16) ? 0 : 2;
    float invT  = 1.0f / (float)(num_t - 1);
    float wbase = 1.0f / (float)num_t;

    float vals[2];
#pragma unroll
    for (int r = 0; r < 2; ++r) {
        int   i = 4 * c + khalf + r;
        float v = 0.0f;
        if (Ncol < 5 && i < num_t) {
            float t = (float)i * invT;
            // ordinary Legendre recurrence up to P8
            float Pl[9];
            Pl[0] = 1.0f; Pl[1] = t;
#pragma unroll
            for (int nn = 1; nn < 8; ++nn)
                Pl[nn + 1] = ((2.0f * nn + 1.0f) * t * Pl[nn]
                              - (float)nn * Pl[nn - 1]) / (float)(nn + 1);
            float w = wbase;
            if (i == 0 || i == num_t - 1) w *= 0.5f;
            float p = (Ncol == 0) ? Pl[0] : (Ncol == 1) ? Pl[2] :
                      (Ncol == 2) ? Pl[4] : (Ncol == 3) ? Pl[6] : Pl[8];
            v = w * p;
        }
        vals[r] = v;
    }
    Bws[tid * 2 + 0] = vals[0];
    Bws[tid * 2 + 1] = vals[1];
}

// ---------------------------------------------------------------------------
// Kernel 2: rho GEMM via V_WMMA_F32_16X16X4_F32, scatter into eigen_values.
// One wave owns a 16-row tile of (n,f,b) rows; 250 WMMAs of K=4 per tile.
// ---------------------------------------------------------------------------
__global__ void __launch_bounds__(256)
k_rho_wmma(const float* __restrict__ shapes, const int* __restrict__ num_t_p,
           const float* __restrict__ Bws, float* __restrict__ eig_out,
           int rows) {
    __shared__ float lds[8][16][17];
    int lane = threadIdx.x & 31;
    int wave = threadIdx.x >> 5;
    int tile = blockIdx.x * 8 + wave;
    int tiles = (rows + 15) >> 4;
    if (tile >= tiles) return;                 // wave-uniform exit

    int   num_t  = *num_t_p;
    int   chunks = (num_t + 3) >> 2;
    float invT   = 1.0f / (float)(num_t - 1);

    // A-fragment row for this lane (both lane halves share the row; they
    // supply different K slots of the same row).
    int rowM = lane & 15;
    int g = tile * 16 + rowM;
    if (g >= rows) g = rows - 1;               // clamp; writes guarded below
    int nidx = g / 9;
    int rem  = g - nidx * 9;
    int f    = rem / 3;
    int bb   = rem - f * 3;
    float bval = (float)(bb + 1);              // BVALS = {1,2,3}

    float4 sp = ((const float4*)shapes)[nidx * 3 + f];
    float D_a = sp.x, D_epar = sp.y, D_eperp = sp.z, zf = sp.w;
    float c1 = bval * D_a;
    float c2 = bval * (D_epar - D_eperp);
    float e0 = (1.0f - zf) * __expf(-bval * D_eperp);
    int khalf = (lane < 16) ? 0 : 2;

    v8f acc = {};
    const v2f* Bv = (const v2f*)Bws;
#pragma unroll 2
    for (int c = 0; c < chunks; ++c) {
        int   i0 = 4 * c + khalf;
        float t0 = (float)i0 * invT;
        float t1 = (float)(i0 + 1) * invT;
        float q0 = t0 * t0, q1 = t1 * t1;
        float a0 = zf * __expf(-c1 * q0) + e0 * __expf(-c2 * q0);
        float a1 = zf * __expf(-c1 * q1) + e0 * __expf(-c2 * q1);
        v2f A; A.x = a0; A.y = a1;
        v2f B = Bv[c * 32 + lane];
        acc = __builtin_amdgcn_wmma_f32_16x16x4_f32(
            /*neg_a=*/false, A, /*neg_b=*/false, B,
            /*c_mod=*/(short)0, acc, /*reuse_a=*/false, /*reuse_b=*/false);
    }

    // D layout: VGPR j -> (M = j + 8*(lane>=16), N = lane%16)
    int rhalf = (lane >> 4) << 3;
    int col   = lane & 15;
#pragma unroll
    for (int j = 0; j < 8; ++j)
        lds[wave][rhalf + j][col] = acc[j];
    // same-wave LDS ops are in order; only this wave reads its slice

    // scatter: eigen_values[n][b][f][k] = rho[row][l_k_idx[k]]
    for (int idx = lane; idx < 16 * 45; idx += 32) {
        int r  = idx / 45;
        int k  = idx - r * 45;
        int gg = tile * 16 + r;
        if (gg < rows) {
            int nn = gg / 9;
            int rr = gg - nn * 9;
            int ff = rr / 3;
            int bq = rr - ff * 3;
            eig_out[(((nn * 3 + bq) * 3 + ff) * 45) + k] = lds[wave][r][c_lidx[k]];
        }
    }
}

// ---------------------------------------------------------------------------
// Kernel 3: real even-degree SH basis (LMAX=8) per (n,f) direction.
// ---------------------------------------------------------------------------
__global__ void k_odf(const float* __restrict__ dirs,
                      float* __restrict__ odf_out, int NF) {
    int tid = blockIdx.x * blockDim.x + threadIdx.x;
    if (tid >= NF) return;
    float x = dirs[tid * 3 + 0], y = dirs[tid * 3 + 1], z = dirs[tid * 3 + 2];
    float rinv = rsqrtf(x * x + y * y + z * z);
    x *= rinv; y *= rinv; z *= rinv;

    // cos/sin of m*phi via Chebyshev recurrence (atan2-free)
    float rxy = sqrtf(x * x + y * y);
    float cphi, sphi;
    if (rxy > 0.0f) { float ir = 1.0f / rxy; cphi = x * ir; sphi = y * ir; }
    else            { cphi = 1.0f; sphi = 0.0f; }
    float cm[9], sm[9];
    cm[0] = 1.0f; sm[0] = 0.0f; cm[1] = cphi; sm[1] = sphi;
#pragma unroll
    for (int m = 1; m < 8; ++m) {
        cm[m + 1] = 2.0f * cphi * cm[m] - cm[m - 1];
        sm[m + 1] = 2.0f * cphi * sm[m] - sm[m - 1];
    }

    // associated Legendre table (fully unrolled -> registers)
    float P[9][9];
    P[0][0] = 1.0f;
    float s = sqrtf(fminf(fmaxf(1.0f - z * z, 0.0f), 1.0f));
#pragma unroll
    for (int m = 1; m <= 8; ++m)
        P[m][m] = -(2.0f * m - 1.0f) * s * P[m - 1][m - 1];
#pragma unroll
    for (int m = 0; m < 8; ++m)
        P[m + 1][m] = (2.0f * m + 1.0f) * z * P[m][m];
#pragma unroll
    for (int m = 0; m <= 8; ++m) {
#pragma unroll
        for (int l = m + 2; l <= 8; ++l)
            P[l][m] = ((2.0f * l - 1.0f) * z * P[l - 1][m]
                       - (float)(l + m - 1) * P[l - 2][m]) / (float)(l - m);
    }

    const float inv4pi = 0.07957747154594767f;
    const float sqrt2  = 1.41421356237309515f;
    float* out = odf_out + (size_t)tid * 45;
    int k = 0;
#pragma unroll
    for (int l = 0; l <= 8; l += 2) {
#pragma unroll
        for (int m = -8; m <= 8; ++m) {
            if (m < -l || m > l) continue;
            int ma = m < 0 ? -m : m;
            float ratio = 1.0f;
#pragma unroll
            for (int i = 1; i <= 16; ++i)
                if (i > l - ma && i <= l + ma) ratio /= (float)i;
            float nrm = sqrtf((2.0f * l + 1.0f) * inv4pi * ratio);
            float v;
            if (m < 0)       v = sqrt2 * nrm * sm[ma] * P[l][ma];
            else if (m == 0) v = nrm * P[l][0];
            else             v = sqrt2 * nrm * cm[ma] * P[l][ma];
            out[k++] = v;
        }
    }
}

// ---------------------------------------------------------------------------
// Kernel 4: Stensor[n,b,k] = sum_f w[n,f] * eigen[n,b,f,k] * odf[n,f,k]
// ---------------------------------------------------------------------------
__global__ void k_stensor(const float* __restrict__ weights,
                          const float* __restrict__ eig,
                          const float* __restrict__ odf,
                          float* __restrict__ s_out, int total) {
    int tid = blockIdx.x * blockDim.x + threadIdx.x;
    if (tid >= total) return;
    int k  = tid % 45;
    int nb = tid / 45;
    int bb = nb % 3;
    int nn = nb / 3;
    float acc = 0.0f;
#pragma unroll
    for (int f = 0; f < 3; ++f)
        acc = fmaf(weights[nn * 3 + f] * odf[(nn * 3 + f) * 45 + k],
                   eig[(((nn * 3 + bb) * 3 + f) * 45) + k], acc);
    s_out[tid] = acc;
}

extern "C" void kernel_launch(void* const* d_in, const int* in_sizes, int n_in,
                              void* d_out, int out_size, void* d_ws, size_t ws_size,
                              hipStream_t stream) {
    const float* directs = (const float*)d_in[0];   // (N,F,3)
    const float* weights = (const float*)d_in[1];   // (N,F)
    const float* shapes  = (const float*)d_in[2];   // (N,F,4)
    const int*   num_t   = (const int*)d_in[3];     // scalar

    int NF   = in_sizes[1];          // N*F
    int N    = NF / 3;
    int rows = NF * 3;               // N*F*B  (B=3 b-values)

    float* out     = (float*)d_out;
    float* s_out   = out;                               // (N,3,45)
    float* odf_out = out + (size_t)N * 3 * 45;          // (N,3,45)
    float* eig_out = out + (size_t)2 * N * 3 * 45;      // (N,3,3,45)
    float* Bws     = (float*)d_ws;                      // B fragments

    // B fragments: supports num_t up to 4096 (1024 chunks x 32 lanes)
    k_precompute_B<<<128, 256, 0, stream>>>(num_t, Bws);

    int tiles   = (rows + 15) / 16;
    int blocks2 = (tiles + 7) / 8;   // 8 waves per block
    k_rho_wmma<<<blocks2, 256, 0, stream>>>(shapes, num_t, Bws, eig_out, rows);

    k_odf<<<(NF + 255) / 256, 256, 0, stream>>>(directs, odf_out, NF);

    int total4 = N * 3 * 45;
    k_stensor<<<(total4 + 255) / 256, 256, 0, stream>>>(weights, eig_out,
                                                        odf_out, s_out, total4);
}